// ScRRAMBLePatching_19164144074963
// MI455X (gfx1250) — compile-verified
//
#include <hip/hip_runtime.h>
#include <hip/hip_bf16.h>

typedef __attribute__((ext_vector_type(2))) float v2f;
typedef __attribute__((ext_vector_type(8))) float v8f;

// Problem (from reference):
//  x: (128, 256, 256, 3) f32   C: (768, 4, 256, 4) f32
//  out: (128, 256, 4, 8, 8) f32 == (b, kl=1024, hw=64)
//  einsum has no shared labels -> out[b,kl,hw] = S_x[b,hw] * S_C[kl]
//    S_C[kl]   = sum_{t<3072} C[t*1024 + kl]
//    S_x[b,hw] = sum over rows r (r%8==h), cols c (c%8==w), 3 channels

// Workspace layout (floats):
//   [0,      32768)  sCp : 32 partials x 1024
//   [32768,  65536)  sXp : 4 partials  x 8192
//   [65536,  66560)  sC  : 1024 finals
//   [66560,  74752)  sX  : 8192 finals

// ---------------------------------------------------------------------------
// Kernel 1: partial C reduction. grid (4, 32): kl = bx*256+tid, part = by.
// Each block sums 96 contiguous-in-t slices; lane-consecutive kl -> coalesced.
// ---------------------------------------------------------------------------
__global__ void scr_reduce_C_part(const float* __restrict__ C, float* __restrict__ sCp) {
    const int kl   = blockIdx.x * blockDim.x + threadIdx.x;   // 0..1023
    const int part = blockIdx.y;                              // 0..31
    const float* p = C + (size_t)part * 96 * 1024 + kl;
    float acc = 0.0f;
#pragma unroll 8
    for (int t = 0; t < 96; ++t) {
        acc += p[(size_t)t * 1024];
    }
    sCp[part * 1024 + kl] = acc;
}

// ---------------------------------------------------------------------------
// Kernel 2: partial x reduction. grid (128, 4): b = bx, part = by, 768 thr.
// Block handles 64 rows [part*64, part*64+64); 64%8==0 so h = j&7 still maps
// correctly. Thread t = col*3+ch walks rows with coalesced 3 KB row reads,
// 8 private accumulators by row%8, then ds_add_f32 into 64 LDS buckets.
// ---------------------------------------------------------------------------
__global__ void scr_reduce_x_part(const float* __restrict__ x, float* __restrict__ sXp) {
    __shared__ float lds[64];
    const int b    = blockIdx.x;
    const int part = blockIdx.y;
    const int t    = threadIdx.x;          // 0..767 : col = t/3, ch = t%3
    const int w    = (t / 3) & 7;          // col % 8

    if (t < 64) lds[t] = 0.0f;
    __syncthreads();

    const float* p = x + (size_t)b * 256 * 768 + (size_t)part * 64 * 768 + t;
    float acc[8] = {0.f, 0.f, 0.f, 0.f, 0.f, 0.f, 0.f, 0.f};
#pragma unroll 8
    for (int j = 0; j < 64; ++j) {
        acc[j & 7] += p[(size_t)j * 768];
    }

#pragma unroll
    for (int h = 0; h < 8; ++h) {
        atomicAdd(&lds[h * 8 + w], acc[h]);   // ds_add_f32
    }
    __syncthreads();

    if (t < 64) sXp[part * 8192 + b * 64 + t] = lds[t];
}

// ---------------------------------------------------------------------------
// Kernel 3: fold partials. 36 blocks x 256 = 9216 threads.
//   idx < 1024          : sC[idx] = sum_{p<32} sCp[p*1024+idx]
//   idx in [1024, 9216) : sX[j]   = sum_{p<4}  sXp[p*8192+j],  j = idx-1024
// ---------------------------------------------------------------------------
__global__ void scr_fold(const float* __restrict__ sCp, const float* __restrict__ sXp,
                         float* __restrict__ sC, float* __restrict__ sX) {
    const int idx = blockIdx.x * blockDim.x + threadIdx.x;
    if (idx < 1024) {
        float acc = 0.0f;
#pragma unroll
        for (int p = 0; p < 32; ++p) acc += sCp[p * 1024 + idx];
        sC[idx] = acc;
    } else {
        const int j = idx - 1024;   // 0..8191
        float acc = 0.0f;
#pragma unroll
        for (int p = 0; p < 4; ++p) acc += sXp[p * 8192 + j];
        sX[j] = acc;
    }
}

// ---------------------------------------------------------------------------
// Kernel 4: rank-1 outer product per image via V_WMMA_F32_16X16X4_F32.
// One wave per 16x16 tile of out[b, kl0:kl0+16, hw0:hw0+16].
//   A[m,k] = S_C[kl0+m] (all k),  B[k,n] = 0.25*S_x[hw0+n] (all k)
//   D[m,n] = 4 * (S_C[m] * 0.25*S_x[n]) == fl(S_C[m]*S_x[n])  (pow2-exact)
// Layout-robust (only needs M/N = lane%16). EXEC all-1s: no divergence.
// D layout: row m = vgpr + 8*(lane>=16), col n = lane%16 -> coalesced stores.
// ---------------------------------------------------------------------------
__global__ void scr_outer_wmma(const float* __restrict__ sC,
                               const float* __restrict__ sX,
                               float* __restrict__ out) {
    const int tid  = threadIdx.x;
    const int lane = tid & 31;
    const int gw   = blockIdx.x * (blockDim.x >> 5) + (tid >> 5); // global wave id
    const int b     = gw >> 8;          // 0..127
    const int rem   = gw & 255;
    const int mtile = rem >> 2;         // 0..63  -> kl0 = mtile*16
    const int ntile = rem & 3;          // 0..3   -> hw0 = ntile*16
    const int l16   = lane & 15;

    const float aval = sC[mtile * 16 + l16];
    const float bval = 0.25f * sX[b * 64 + ntile * 16 + l16];

    v2f A  = {aval, aval};
    v2f Bv = {bval, bval};
    v8f Cacc = {};
    // 8 args: (neg_a, A, neg_b, B, c_mod, C, reuse_a, reuse_b)
    v8f D = __builtin_amdgcn_wmma_f32_16x16x4_f32(
        false, A, false, Bv, (short)0, Cacc, false, false);

    const int mhi = (lane >> 4) << 3;   // 0 or 8
    float* basep = out + ((size_t)b * 1024 + (size_t)mtile * 16 + mhi) * 64
                       + ntile * 16 + l16;
#pragma unroll
    for (int v = 0; v < 8; ++v) {
        basep[(size_t)v * 64] = D[v];   // row kl0+mhi+v, contiguous over l16
    }
}

// ---------------------------------------------------------------------------
extern "C" void kernel_launch(void* const* d_in, const int* in_sizes, int n_in,
                              void* d_out, int out_size, void* d_ws, size_t ws_size,
                              hipStream_t stream) {
    (void)in_sizes; (void)n_in; (void)out_size; (void)ws_size;

    const float* x = (const float*)d_in[0];   // 128*256*256*3
    const float* C = (const float*)d_in[1];   // 768*4*256*4
    float* out = (float*)d_out;               // 128*1024*64

    float* sCp = (float*)d_ws;                // 32*1024
    float* sXp = sCp + 32 * 1024;             // 4*8192
    float* sC  = sXp + 4 * 8192;              // 1024
    float* sX  = sC + 1024;                   // 8192

    scr_reduce_C_part<<<dim3(4, 32), 256, 0, stream>>>(C, sCp);
    scr_reduce_x_part<<<dim3(128, 4), 768, 0, stream>>>(x, sXp);
    scr_fold<<<36, 256, 0, stream>>>(sCp, sXp, sC, sX);
    // 32768 tiles / 8 waves per 256-thread block = 4096 blocks (exact cover)
    scr_outer_wmma<<<4096, 256, 0, stream>>>(sC, sX, out);
}